// SampleChamfer_67740224192625
// MI455X (gfx1250) — compile-verified
//
#include <hip/hip_runtime.h>
#include <hip/hip_bf16.h>

typedef __attribute__((ext_vector_type(2))) float v2f;
typedef __attribute__((ext_vector_type(8))) float v8f;

#define NPTS   16384      // N_POINTS
#define NSAMP  8192       // N_SAMPLES
#define NTILE  (NSAMP/16) // 512 b-tiles / 512 a-tiles

// Workspace layout (floats):
//   [0,      32768)  Amat : b-point rows  (x, y, z, |b|^2)      128 KB
//   [32768,  65536)  Bmat : a-point cols  (-2x, -2y, -2z, 1)    128 KB
//   [65536,  73728)  sqA  : |a|^2 per sampled a-point            32 KB
//   [73728,  74240)  partial per-tile sums (512)                  2 KB

__global__ void chamfer_prep(const float* __restrict__ a,
                             const float* __restrict__ b,
                             const int*   __restrict__ a_idx,
                             const int*   __restrict__ b_idx,
                             float* __restrict__ Amat,
                             float* __restrict__ Bmat,
                             float* __restrict__ sqA) {
    int i = blockIdx.x * blockDim.x + threadIdx.x;
    if (i >= NSAMP) return;

    int ai = a_idx[i];
    float ax = a[0 * NPTS + ai];
    float ay = a[1 * NPTS + ai];
    float az = a[2 * NPTS + ai];
    Bmat[i * 4 + 0] = -2.0f * ax;
    Bmat[i * 4 + 1] = -2.0f * ay;
    Bmat[i * 4 + 2] = -2.0f * az;
    Bmat[i * 4 + 3] = 1.0f;
    sqA[i] = ax * ax + ay * ay + az * az;

    int bi = b_idx[i];
    float bx = b[0 * NPTS + bi];
    float by = b[1 * NPTS + bi];
    float bz = b[2 * NPTS + bi];
    Amat[i * 4 + 0] = bx;
    Amat[i * 4 + 1] = by;
    Amat[i * 4 + 2] = bz;
    Amat[i * 4 + 3] = bx * bx + by * by + bz * bz;
}

// One wave32 per tile of 16 a-points. B operand (a-points, with -2 and the
// +1 column for |b|^2) stays resident in 2 VGPRs for the whole b-loop.
// Each iteration: one 8B coalesced load of the A tile slice, one
// v_wmma_f32_16x16x4_f32 producing D[m][n] = |pb_m|^2 - 2 pb_m . pa_n,
// then 8 v_min to keep the running column minimum.
__global__ void __launch_bounds__(256)
chamfer_wmma(const float* __restrict__ Amat,
             const float* __restrict__ Bmat,
             const float* __restrict__ sqA,
             float* __restrict__ partial) {
    const int lane = threadIdx.x & 31;
    const int wave = threadIdx.x >> 5;
    const int tile = blockIdx.x * (blockDim.x >> 5) + wave;  // 0..511
    const int half = lane >> 4;   // K pair select: {0,1} vs {2,3}
    const int nl   = lane & 15;   // column / row within 16-tile
    const int n    = tile * 16 + nl;

    // B matrix 4x16: lane holds column n, K = half*2 + {0,1}
    v2f breg = *(const v2f*)(Bmat + n * 4 + half * 2);

    v8f mn;
#pragma unroll
    for (int r = 0; r < 8; ++r) mn[r] = 3.402823466e38f;

    // A matrix 16x4: lane holds row m = nl of the current b-tile, same K pair
    const float* ap = Amat + nl * 4 + half * 2;
    for (int jt = 0; jt < NTILE; ++jt) {
        v2f areg = *(const v2f*)(ap + jt * 64);          // 16 rows * 4 floats
        __builtin_prefetch(ap + (jt + 8) * 64, 0, 0);    // global_prefetch_b8
        v8f c = {0.f, 0.f, 0.f, 0.f, 0.f, 0.f, 0.f, 0.f};
        v8f d = __builtin_amdgcn_wmma_f32_16x16x4_f32(
            /*neg_a=*/false, areg, /*neg_b=*/false, breg,
            /*c_mod=*/(short)0, c, /*reuse_a=*/false, /*reuse_b=*/false);
#pragma unroll
        for (int r = 0; r < 8; ++r) mn[r] = fminf(mn[r], d[r]);
    }

    // D layout: VGPR r -> M = r (lanes 0-15) or M = r+8 (lanes 16-31),
    // N = lane%16. Reduce min over the 8 VGPRs, then across the two halves.
    float m = fminf(fminf(fminf(mn[0], mn[1]), fminf(mn[2], mn[3])),
                    fminf(fminf(mn[4], mn[5]), fminf(mn[6], mn[7])));
    m = fminf(m, __shfl_xor(m, 16, 32));

    // Per-column contribution: min_m D[m][n] + |pa_n|^2. Upper half zeroed
    // to avoid double counting; full-wave butterfly sum, lane 0 writes.
    float val = half ? 0.0f : (m + sqA[n]);
#pragma unroll
    for (int s = 1; s < 32; s <<= 1) val += __shfl_xor(val, s, 32);
    if (lane == 0) partial[tile] = val;
}

// Deterministic final reduction of 512 partials by a single wave.
__global__ void chamfer_reduce(const float* __restrict__ partial,
                               float* __restrict__ out) {
    int lane = threadIdx.x & 31;
    float s = 0.0f;
    for (int k = lane; k < NTILE; k += 32) s += partial[k];
#pragma unroll
    for (int d = 1; d < 32; d <<= 1) s += __shfl_xor(s, d, 32);
    if (lane == 0) out[0] = s;
}

extern "C" void kernel_launch(void* const* d_in, const int* in_sizes, int n_in,
                              void* d_out, int out_size, void* d_ws, size_t ws_size,
                              hipStream_t stream) {
    const float* a     = (const float*)d_in[0];
    const float* b     = (const float*)d_in[1];
    const int*   a_idx = (const int*)d_in[2];
    const int*   b_idx = (const int*)d_in[3];
    float* out = (float*)d_out;

    float* ws      = (float*)d_ws;
    float* Amat    = ws;                    // 32768 floats
    float* Bmat    = ws + NSAMP * 4;        // 32768 floats
    float* sqA     = ws + NSAMP * 8;        //  8192 floats
    float* partial = ws + NSAMP * 9;        //   512 floats

    chamfer_prep<<<(NSAMP + 255) / 256, 256, 0, stream>>>(
        a, b, a_idx, b_idx, Amat, Bmat, sqA);

    // 512 tiles, 8 waves per 256-thread block -> 64 blocks
    chamfer_wmma<<<NTILE / 8, 256, 0, stream>>>(Amat, Bmat, sqA, partial);

    chamfer_reduce<<<1, 32, 0, stream>>>(partial, out);
}